// RegularizationLossFn_34806414967010
// MI455X (gfx1250) — compile-verified
//
#include <hip/hip_runtime.h>

#define B_DIM 256
#define K_DIM 32
#define L_DIM 4096

constexpr float SIGMA        = 1.5f;
constexpr float EPS          = 1e-5f;
constexpr float PEAK_LAMBDA  = 0.005f;
constexpr float ORTHO_LAMBDA = 20000.0f;

typedef __attribute__((ext_vector_type(2))) float v2f;
typedef __attribute__((ext_vector_type(8))) float v8f;

// ---------------------------------------------------------------------------
// Kernel 1: peak (KL) loss partials, one block per (b,k) row of length 4096.
// 256 threads, each holds 16 elements in registers (coalesced stride-256).
// ---------------------------------------------------------------------------
__global__ __launch_bounds__(256) void peak_kernel(const float* __restrict__ a,
                                                   float* __restrict__ kl_out) {
    const int row = blockIdx.x;                       // b*K + k, 0..8191
    const float* __restrict__ p = a + (size_t)row * L_DIM;
    const int tid = threadIdx.x;

    __shared__ float sv[256];
    __shared__ int   si[256];

    // ---- load row slice into registers ----
    float av[16];
#pragma unroll
    for (int j = 0; j < 16; ++j) av[j] = p[tid + 256 * j];

    // ---- argmax (first occurrence of max, matching jnp.argmax) ----
    float bestv = av[0];
    int   besti = tid;
#pragma unroll
    for (int j = 1; j < 16; ++j) {
        const int l = tid + 256 * j;
        if (av[j] > bestv) { bestv = av[j]; besti = l; }
    }
    sv[tid] = bestv; si[tid] = besti;
    __syncthreads();
    for (int s = 128; s > 0; s >>= 1) {
        if (tid < s) {
            const float v2 = sv[tid + s]; const int i2 = si[tid + s];
            const float v1 = sv[tid];     const int i1 = si[tid];
            if (v2 > v1 || (v2 == v1 && i2 < i1)) { sv[tid] = v2; si[tid] = i2; }
        }
        __syncthreads();
    }
    const int idx = si[0];

    // ---- Gaussian target: g = exp(-0.5*((l-idx)/sigma)^2), sum over L ----
    float e[16];
    float sg = 0.0f;
#pragma unroll
    for (int j = 0; j < 16; ++j) {
        const float d = (float)(tid + 256 * j - idx) * (1.0f / SIGMA);
        const float g = expf(-0.5f * d * d);
        e[j] = g;
        sg += g;
    }
    __syncthreads();
    sv[tid] = sg;
    __syncthreads();
    for (int s = 128; s > 0; s >>= 1) {
        if (tid < s) sv[tid] += sv[tid + s];
        __syncthreads();
    }
    const float ginv = 1.0f / sv[0];
    __syncthreads();

    // ---- KL: sum t*(log t - log p), t = g/sum_g + eps, p = a + eps ----
    float kl = 0.0f;
#pragma unroll
    for (int j = 0; j < 16; ++j) {
        const float t  = e[j] * ginv + EPS;
        const float pv = av[j] + EPS;
        kl += t * (logf(t) - logf(pv));
    }
    sv[tid] = kl;
    __syncthreads();
    for (int s = 128; s > 0; s >>= 1) {
        if (tid < s) sv[tid] += sv[tid + s];
        __syncthreads();
    }
    if (tid == 0) kl_out[row] = sv[0];
}

// ---------------------------------------------------------------------------
// Kernel 2: ortho loss partials. One block (4 waves / 128 threads) per batch.
// Wave w computes 16x16 tile (ti,tj) of dot = P*P^T (P = a + eps, 32 x 4096)
// using V_WMMA_F32_16X16X4_F32, marching the K dimension over L in chunks of
// 4 (two accumulators to break the WMMA dependency chain). Then reduces the
// sum of squared off-diagonal entries.
// ---------------------------------------------------------------------------
__global__ __launch_bounds__(128) void ortho_kernel(const float* __restrict__ a,
                                                    float* __restrict__ ortho_out) {
    const int b = blockIdx.x;                         // 0..255
    const float* __restrict__ P = a + (size_t)b * (K_DIM * L_DIM);

    const int lane = threadIdx.x & 31;
    const int wave = threadIdx.x >> 5;                // 0..3
    const int ti = wave >> 1;                         // row tile (0/1)
    const int tj = wave & 1;                          // col tile (0/1)

    // f32 16x16x4 frag layout: lane holds row = tile*16 + (lane&15),
    // VGPR0 = K (lane<16 ? 0 : 2), VGPR1 = K+1  -> one aligned float2 load.
    const int rA    = ti * 16 + (lane & 15);
    const int rB    = tj * 16 + (lane & 15);
    const int kbase = (lane >> 4) << 1;               // 0 or 2

    const float* __restrict__ pA = P + (size_t)rA * L_DIM + kbase;
    const float* __restrict__ pB = P + (size_t)rB * L_DIM + kbase;

    const v2f eps2 = {EPS, EPS};
    v8f c0 = {};
    v8f c1 = {};

    for (int k = 0; k < L_DIM; k += 8) {
        v2f a0 = *(const v2f*)(pA + k)     + eps2;
        v2f b0 = *(const v2f*)(pB + k)     + eps2;
        v2f a1 = *(const v2f*)(pA + k + 4) + eps2;
        v2f b1 = *(const v2f*)(pB + k + 4) + eps2;
        // D = A(16x4) x B(4x16) + C : v_wmma_f32_16x16x4_f32
        c0 = __builtin_amdgcn_wmma_f32_16x16x4_f32(
                 false, a0, false, b0, (short)0, c0, false, false);
        c1 = __builtin_amdgcn_wmma_f32_16x16x4_f32(
                 false, a1, false, b1, (short)0, c1, false, false);
    }

    // C/D layout: VGPR r -> M = r + (lane<16 ? 0 : 8), N = lane & 15.
    const int n    = lane & 15;
    const int mofs = (lane >> 4) * 8;
    float ss = 0.0f;
#pragma unroll
    for (int r = 0; r < 8; ++r) {
        const float v   = c0[r] + c1[r];
        const int   m   = r + mofs;
        const bool  dia = (ti == tj) && (m == n);
        ss += dia ? 0.0f : v * v;
    }

    __shared__ float red[128];
    red[threadIdx.x] = ss;
    __syncthreads();
    for (int s = 64; s > 0; s >>= 1) {
        if (threadIdx.x < s) red[threadIdx.x] += red[threadIdx.x + s];
        __syncthreads();
    }
    if (threadIdx.x == 0) ortho_out[b] = red[0];
}

// ---------------------------------------------------------------------------
// Kernel 3: deterministic final reduction -> (loss, peak_loss, ortho_loss)
// ---------------------------------------------------------------------------
__global__ __launch_bounds__(256) void finalize_kernel(const float* __restrict__ kl,
                                                       const float* __restrict__ ortho,
                                                       float* __restrict__ out) {
    __shared__ float s1[256];
    __shared__ float s2[256];
    const int tid = threadIdx.x;

    float accA = 0.0f;
    for (int i = tid; i < B_DIM * K_DIM; i += 256) accA += kl[i];
    float accB = (tid < B_DIM) ? ortho[tid] : 0.0f;

    s1[tid] = accA; s2[tid] = accB;
    __syncthreads();
    for (int s = 128; s > 0; s >>= 1) {
        if (tid < s) { s1[tid] += s1[tid + s]; s2[tid] += s2[tid + s]; }
        __syncthreads();
    }
    if (tid == 0) {
        const float peak = PEAK_LAMBDA * s1[0];
        const float orth = ORTHO_LAMBDA * (s2[0] / (float)(B_DIM * K_DIM * K_DIM));
        out[0] = peak + orth;   // loss
        out[1] = peak;          // peak_loss
        out[2] = orth;          // ortho_loss
    }
}

extern "C" void kernel_launch(void* const* d_in, const int* in_sizes, int n_in,
                              void* d_out, int out_size, void* d_ws, size_t ws_size,
                              hipStream_t stream) {
    const float* attns = (const float*)d_in[0];     // [1, B, K, L] fp32
    float* kl_ws    = (float*)d_ws;                 // 8192 floats
    float* ortho_ws = kl_ws + B_DIM * K_DIM;        // 256 floats
    float* out      = (float*)d_out;                // 3 floats

    peak_kernel<<<B_DIM * K_DIM, 256, 0, stream>>>(attns, kl_ws);
    ortho_kernel<<<B_DIM, 128, 0, stream>>>(attns, ortho_ws);
    finalize_kernel<<<1, 256, 0, stream>>>(kl_ws, ortho_ws, out);
}